// LorentzTransformerEncoder_90632399880641
// MI455X (gfx1250) — compile-verified
//
#include <hip/hip_runtime.h>
#include <math.h>

typedef float v2f __attribute__((ext_vector_type(2)));
typedef float v8f __attribute__((ext_vector_type(8)));

#define T4  4096      // total tokens (b*n)
#define DD  768       // spatial dim
#define HH  769       // hidden with time
#define KP1 772       // 769 padded to mult of 4
#define DM  3072      // mlp spatial dim
#define KP2 3076      // 3073 padded to mult of 4
#define NHD 12        // heads
#define HD  64        // head dim

// ---------------- WMMA helpers (fp32 16x16x4) ----------------
__device__ __forceinline__ v8f wmma4(v2f a, v2f b, v8f c) {
  return __builtin_amdgcn_wmma_f32_16x16x4_f32(false, a, false, b, (short)0, c, false, false);
}

// Fragment load for A or B from a K-major matrix [K][stride]:
// lanes 0-15 hold (row n0+lane, K=k..k+1), lanes 16-31 hold (row n0+lane-16, K=k+2..k+3)
__device__ __forceinline__ v2f ldFrag(const float* __restrict__ base, int stride,
                                      int k, int n0, int lane) {
  int ln   = lane & 15;
  int koff = (lane >> 4) << 1;
  const float* p = base + (size_t)(k + koff) * stride + n0 + ln;
  v2f r; r.x = p[0]; r.y = p[stride];
  return r;
}

// ---------------- weight transpose+pad: W[F][Korig] -> WT[Kpad][F] ----------------
__global__ void transposeW(const float* __restrict__ W, float* __restrict__ WT,
                           int Korig, int Kpad, int F) {
  int idx = blockIdx.x * blockDim.x + threadIdx.x;
  if (idx >= Kpad * F) return;
  int k = idx / F, f = idx - k * F;
  WT[idx] = (k < Korig) ? W[(size_t)f * Korig + k] : 0.0f;
}

// ---------------- LN1: x token-major [t][769] -> xlnT K-major [772][4096] ----------------
__global__ void ln1_kernel(const float* __restrict__ x, const float* __restrict__ g,
                           const float* __restrict__ bb, float* __restrict__ xlnT) {
  int t = blockIdx.x * blockDim.x + threadIdx.x;
  if (t >= T4) return;
  const float* xr = x + (size_t)t * HH + 1;
  float s = 0.f, s2 = 0.f;
  for (int f = 0; f < DD; ++f) { float v = xr[f]; s += v; s2 += v * v; }
  float mu   = s  * (1.0f / DD);
  float var  = s2 * (1.0f / DD) - mu * mu;
  float rstd = 1.0f / sqrtf(var + 1e-5f);
  float acc = 0.f;
  for (int f = 0; f < DD; ++f) {
    float v = (xr[f] - mu) * rstd * g[f] + bb[f];
    xlnT[(size_t)(1 + f) * T4 + t] = v;
    acc += v * v;
  }
  xlnT[t] = sqrtf(1.0f + acc);               // time row
  xlnT[(size_t)769 * T4 + t] = 0.f;          // K padding
  xlnT[(size_t)770 * T4 + t] = 0.f;
  xlnT[(size_t)771 * T4 + t] = 0.f;
}

// ---------------- LN2: inT K-major [768+][4096] -> hT [772][4096] ----------------
__global__ void ln2_kernel(const float* __restrict__ inT, const float* __restrict__ g,
                           const float* __restrict__ bb, float* __restrict__ hT) {
  int t = blockIdx.x * blockDim.x + threadIdx.x;
  if (t >= T4) return;
  float s = 0.f, s2 = 0.f;
  for (int f = 0; f < DD; ++f) { float v = inT[(size_t)f * T4 + t]; s += v; s2 += v * v; }
  float mu   = s  * (1.0f / DD);
  float var  = s2 * (1.0f / DD) - mu * mu;
  float rstd = 1.0f / sqrtf(var + 1e-5f);
  float acc = 0.f;
  for (int f = 0; f < DD; ++f) {
    float v = (inT[(size_t)f * T4 + t] - mu) * rstd * g[f] + bb[f];
    hT[(size_t)(1 + f) * T4 + t] = v;
    acc += v * v;
  }
  hT[t] = sqrtf(1.0f + acc);
  hT[(size_t)769 * T4 + t] = 0.f;
  hT[(size_t)770 * T4 + t] = 0.f;
  hT[(size_t)771 * T4 + t] = 0.f;
}

// ---------------- generic WMMA GEMM: OutT[f][t] = sum_k WT[k][f] * XT[k][t] ----------------
// mode 0: plain   1: exact gelu   2: + extra[t*769+1+f] (token-major x residual)
// mode 3: + extra[f*4096+t] (feature-major residual)
__global__ void gemm_kt(const float* __restrict__ WT, const float* __restrict__ XT,
                        float* __restrict__ OutT, int K, int F,
                        int mode, const float* __restrict__ extra) {
  int lane = threadIdx.x & 31;
  int wave = threadIdx.x >> 5;
  int f0 = (blockIdx.y * 8 + wave) * 16;
  int t0 = blockIdx.x * 64;
  v8f a0 = {}, a1 = {}, a2 = {}, a3 = {};
  for (int k = 0; k < K; k += 4) {
    v2f af = ldFrag(WT, F, k, f0, lane);
    v2f b0 = ldFrag(XT, T4, k, t0,      lane);
    v2f b1 = ldFrag(XT, T4, k, t0 + 16, lane);
    v2f b2 = ldFrag(XT, T4, k, t0 + 32, lane);
    v2f b3 = ldFrag(XT, T4, k, t0 + 48, lane);
    a0 = wmma4(af, b0, a0);
    a1 = wmma4(af, b1, a1);
    a2 = wmma4(af, b2, a2);
    a3 = wmma4(af, b3, a3);
  }
  int laneN = lane & 15;
  int half8 = (lane >> 4) * 8;
#pragma unroll
  for (int r = 0; r < 8; ++r) {
    int f = f0 + r + half8;
    float vv[4] = { a0[r], a1[r], a2[r], a3[r] };
#pragma unroll
    for (int c = 0; c < 4; ++c) {
      int t = t0 + c * 16 + laneN;
      float v = vv[c];
      if (mode == 1)      v = 0.5f * v * (1.0f + erff(v * 0.70710678118654752f));
      else if (mode == 2) v += extra[(size_t)t * HH + 1 + f];
      else if (mode == 3) v += extra[(size_t)f * T4 + t];
      OutT[(size_t)f * T4 + t] = v;
    }
  }
}

// ---------------- per-head prep: q_t, k_t, v_tan = logmap0(add_time(v)) ----------------
__global__ void head_prep(const float* __restrict__ qT, const float* __restrict__ kT,
                          const float* __restrict__ vT, float* __restrict__ qt_,
                          float* __restrict__ kt_, float* __restrict__ vtanJ) {
  int idx = blockIdx.x * blockDim.x + threadIdx.x;   // h*4096 + token
  if (idx >= NHD * T4) return;
  int h = idx >> 12;
  int t = idx & (T4 - 1);
  const float* qb = qT + (size_t)h * HD * T4 + t;
  const float* kb = kT + (size_t)h * HD * T4 + t;
  const float* vb = vT + (size_t)h * HD * T4 + t;
  float sq = 0.f, sk = 0.f, sv = 0.f;
  for (int d = 0; d < HD; ++d) {
    float q = qb[(size_t)d * T4]; sq += q * q;
    float k = kb[(size_t)d * T4]; sk += k * k;
    float v = vb[(size_t)d * T4]; sv += v * v;
  }
  qt_[idx] = sqrtf(1.0f + sq);
  kt_[idx] = sqrtf(1.0f + sk);
  float vt   = fmaxf(sqrtf(1.0f + sv), 1.0f + 1e-7f);
  float dist = logf(vt + sqrtf(vt * vt - 1.0f));           // arccosh
  float fac  = dist / fmaxf(sqrtf(sv), 1e-8f);
  int b = t >> 10, n = t & 1023;
  float* vout = vtanJ + (((size_t)b * NHD + h) * 1024 + n) * HD;
  for (int d = 0; d < HD; ++d) vout[d] = vb[(size_t)d * T4] * fac;
}

// ---------------- attention: flash-style per (b,h), one 16-row i-tile per wave ----------------
__global__ void attn_kernel(const float* __restrict__ qT, const float* __restrict__ kT,
                            const float* __restrict__ qt_, const float* __restrict__ kt_,
                            const float* __restrict__ vtanJ, float* __restrict__ yT,
                            float* __restrict__ tH,
                            const float* __restrict__ araw, const float* __restrict__ traw,
                            const float* __restrict__ lraw) {
  __shared__ float Plds[8 * 256];
  int lane  = threadIdx.x & 31;
  int wave  = threadIdx.x >> 5;
  int bh    = blockIdx.y;
  int b     = bh / NHD, h = bh - b * NHD;
  int tb    = b * 1024;
  int i0    = (blockIdx.x * 8 + wave) * 16;
  int laneN = lane & 15;
  int half8 = (lane >> 4) * 8;
  int koff  = (lane >> 4) * 2;

  float alpha  = log1pf(expf(araw[0]));   // softplus
  float tau    = log1pf(expf(traw[0]));
  float lam    = log1pf(expf(lraw[0]));
  float spnegm = log1pf(expf(-0.1f));

  const float* qbase = qT + (size_t)h * HD * T4 + tb;
  const float* kbase = kT + (size_t)h * HD * T4 + tb;

  // preload 16x64 Q tile as 16 A-fragments (reused over all j)
  v2f qa[16];
#pragma unroll
  for (int ks = 0; ks < 16; ++ks) {
    const float* p = qbase + (size_t)(ks * 4 + koff) * T4 + i0 + laneN;
    qa[ks].x = p[0]; qa[ks].y = p[T4];
  }

  float qtv[8], coshOQ[8], sOQ[8], Brow[8];
#pragma unroll
  for (int r = 0; r < 8; ++r) {
    float qt = qt_[(size_t)h * T4 + tb + i0 + r + half8];
    qtv[r] = qt;
    float cq = fmaxf(qt, 1.0f + 1e-7f);
    coshOQ[r] = cq;
    float ct = fminf(logf(cq + sqrtf(cq * cq - 1.0f)), 40.0f);
    sOQ[r]  = sinhf(ct);
    Brow[r] = alpha * ct / (1.0f + alpha * ct);
  }

  float m[8], l[8];
  v8f o0 = {}, o1 = {}, o2 = {}, o3 = {};
#pragma unroll
  for (int r = 0; r < 8; ++r) { m[r] = -1e30f; l[r] = 0.f; }

  float* pl = &Plds[wave * 256];

#pragma unroll 1
  for (int j = 0; j < 64; ++j) {
    int j0 = j * 16;
    float ktv    = kt_[(size_t)h * T4 + tb + j0 + laneN];
    float coshOK = fmaxf(ktv, 1.0f + 1e-7f);

    // S = q_s k_s^T  (K=64 -> 16 wmma)
    v8f s = {};
#pragma unroll
    for (int ks = 0; ks < 16; ++ks) {
      v2f bf;
      const float* p = kbase + (size_t)(ks * 4 + koff) * T4 + j0 + laneN;
      bf.x = p[0]; bf.y = p[T4];
      s = wmma4(qa[ks], bf, s);
    }

    float lg[8];
#pragma unroll
    for (int r = 0; r < 8; ++r) {
      float c    = fmaxf(qtv[r] * ktv - s[r], 1.0f);         // raw_cosh
      bool selfp = c < 1.0f + 1e-5f;
      float cs   = selfp ? 2.0f : c;
      float sq1  = sqrtf(cs * cs - 1.0f);                    // sinh(arccosh(cs))
      float dist = logf(cs + sq1);                           // arccosh(cs)
      float numer = c * coshOQ[r] - coshOK;
      float Z    = selfp ? 1.0f
                         : fminf(fmaxf(numer / (sOQ[r] * sq1), -1.0f), 1.0f);
      float dl   = fminf(dist, 40.0f);
      float pen  = -lam * log1pf(dl * dl);
      float entail = log1pf(expf(Brow[r] + Z - 0.1f)) - spnegm;
      lg[r] = pen - tau * entail;
    }

    // online softmax per row (rows live in 16-lane groups)
#pragma unroll
    for (int r = 0; r < 8; ++r) {
      float v = lg[r];
#pragma unroll
      for (int msk = 1; msk < 16; msk <<= 1) v = fmaxf(v, __shfl_xor(v, msk, 16));
      float nm = fmaxf(m[r], v);
      float sc = expf(m[r] - nm);
      float p  = expf(lg[r] - nm);
      float rs = p;
#pragma unroll
      for (int msk = 1; msk < 16; msk <<= 1) rs += __shfl_xor(rs, msk, 16);
      l[r] = l[r] * sc + rs;
      m[r] = nm;
      o0[r] *= sc; o1[r] *= sc; o2[r] *= sc; o3[r] *= sc;
      pl[(r + half8) * 16 + laneN] = p;   // D-layout -> LDS
    }

    asm volatile("s_wait_dscnt 0" ::: "memory");

    // O += P @ Vtan  (A=P from LDS, B=vtan token-major [j][d])
    const float* vb = vtanJ + (((size_t)bh) * 1024 + j0) * HD;
#pragma unroll
    for (int ks = 0; ks < 4; ++ks) {
      int kk = ks * 4 + koff;
      v2f pa;
      pa.x = pl[laneN * 16 + kk];
      pa.y = pl[laneN * 16 + kk + 1];
#pragma unroll
      for (int nd = 0; nd < 4; ++nd) {
        v2f bv;
        const float* bp = vb + (size_t)kk * HD + nd * 16 + laneN;
        bv.x = bp[0]; bv.y = bp[HD];
        if      (nd == 0) o0 = wmma4(pa, bv, o0);
        else if (nd == 1) o1 = wmma4(pa, bv, o1);
        else if (nd == 2) o2 = wmma4(pa, bv, o2);
        else              o3 = wmma4(pa, bv, o3);
      }
    }
    asm volatile("s_wait_dscnt 0" ::: "memory");
  }

  // finalize: normalize, expmap0, scatter into yT (feature-major) + per-head time
  float inv8 = 1.0f / (1.0f + 1e-8f);
#pragma unroll
  for (int r = 0; r < 8; ++r) {
    float inv = inv8 / fmaxf(l[r], 1e-30f);
    float u0 = o0[r] * inv, u1 = o1[r] * inv, u2 = o2[r] * inv, u3 = o3[r] * inv;
    float nn = u0 * u0 + u1 * u1 + u2 * u2 + u3 * u3;
#pragma unroll
    for (int msk = 1; msk < 16; msk <<= 1) nn += __shfl_xor(nn, msk, 16);
    float n_ = sqrtf(nn);
    float th = coshf(n_);
    float sf = sinhf(n_) / fmaxf(n_, 1e-8f);
    size_t tok = (size_t)tb + i0 + r + half8;
    if (laneN == 0) tH[(size_t)h * T4 + tok] = th;
    yT[(size_t)(1 + h * HD +  0 + laneN) * T4 + tok] = sf * u0;
    yT[(size_t)(1 + h * HD + 16 + laneN) * T4 + tok] = sf * u1;
    yT[(size_t)(1 + h * HD + 32 + laneN) * T4 + tok] = sf * u2;
    yT[(size_t)(1 + h * HD + 48 + laneN) * T4 + tok] = sf * u3;
  }
}

// ---------------- merge head times: t_new = sqrt(sum_h t^2 - (h-1)K) ----------------
__global__ void tnew_kernel(const float* __restrict__ tH, float* __restrict__ yT) {
  int t = blockIdx.x * blockDim.x + threadIdx.x;
  if (t >= T4) return;
  float s = 0.f;
  for (int h = 0; h < NHD; ++h) { float a = tH[(size_t)h * T4 + t]; s += a * a; }
  yT[t] = sqrtf(fmaxf(s - (float)(NHD - 1), 0.0f));
  yT[(size_t)769 * T4 + t] = 0.f;
  yT[(size_t)770 * T4 + t] = 0.f;
  yT[(size_t)771 * T4 + t] = 0.f;
}

// ---------------- MLP hidden time row + padding ----------------
__global__ void mtime_kernel(float* __restrict__ mT) {
  int t = blockIdx.x * blockDim.x + threadIdx.x;
  if (t >= T4) return;
  float acc = 0.f;
  for (int f = 0; f < DM; ++f) { float v = mT[(size_t)(1 + f) * T4 + t]; acc += v * v; }
  mT[t] = sqrtf(1.0f + acc);
  mT[(size_t)3073 * T4 + t] = 0.f;
  mT[(size_t)3074 * T4 + t] = 0.f;
  mT[(size_t)3075 * T4 + t] = 0.f;
}

// ---------------- final add_time -> token-major output [t][769] ----------------
__global__ void final_kernel(const float* __restrict__ finT, float* __restrict__ out) {
  int t = blockIdx.x * blockDim.x + threadIdx.x;
  if (t >= T4) return;
  float acc = 0.f;
  float* orow = out + (size_t)t * HH;
  for (int f = 0; f < DD; ++f) {
    float v = finT[(size_t)f * T4 + t];
    orow[1 + f] = v;
    acc += v * v;
  }
  orow[0] = sqrtf(1.0f + acc);
}

// ---------------- host launch ----------------
extern "C" void kernel_launch(void* const* d_in, const int* in_sizes, int n_in,
                              void* d_out, int out_size, void* d_ws, size_t ws_size,
                              hipStream_t stream) {
  (void)in_sizes; (void)n_in; (void)out_size; (void)ws_size;
  const float* x   = (const float*)d_in[0];
  const float* Wq  = (const float*)d_in[1];
  const float* Wk  = (const float*)d_in[2];
  const float* Wv  = (const float*)d_in[3];
  const float* Wo  = (const float*)d_in[4];
  const float* g1  = (const float*)d_in[5];
  const float* b1  = (const float*)d_in[6];
  const float* g2  = (const float*)d_in[7];
  const float* b2  = (const float*)d_in[8];
  const float* Wm1 = (const float*)d_in[9];
  const float* Wm2 = (const float*)d_in[10];
  const float* ar  = (const float*)d_in[11];
  const float* tr  = (const float*)d_in[12];
  const float* lr  = (const float*)d_in[13];
  float* out = (float*)d_out;
  float* W   = (float*)d_ws;

  const size_t SZ_W  = (size_t)KP1 * DD;        // 592,896
  const size_t SZ_M1 = (size_t)KP1 * DM;        // 2,371,584
  const size_t SZ_M2 = (size_t)KP2 * DD;        // 2,362,368
  const size_t SZ_A7 = (size_t)KP1 * T4;        // 3,162,112
  const size_t SZ_AD = (size_t)DD  * T4;        // 3,145,728
  const size_t SZ_BIG= (size_t)KP2 * T4;        // 12,599,296
  const size_t SZ_HT = (size_t)NHD * T4;        // 49,152

  float* WTq  = W;
  float* WTk  = WTq  + SZ_W;
  float* WTv  = WTk  + SZ_W;
  float* WTo  = WTv  + SZ_W;
  float* WTm1 = WTo  + SZ_W;
  float* WTm2 = WTm1 + SZ_M1;
  float* xlnT = WTm2 + SZ_M2;     // reused later as finT
  float* BIG  = xlnT + SZ_A7;     // q/k/v/vtan during attention; mT during MLP
  float* qT    = BIG;
  float* kT    = qT + SZ_AD;
  float* vT    = kT + SZ_AD;
  float* vtanJ = vT + SZ_AD;
  float* mT    = BIG;
  float* yT    = BIG  + SZ_BIG;
  float* outT  = yT   + SZ_A7;
  float* hT    = outT + SZ_AD;
  float* qt_   = hT   + SZ_A7;
  float* kt_   = qt_  + SZ_HT;
  float* tH    = kt_  + SZ_HT;
  float* finT  = xlnT;

  dim3 blk(256);
  dim3 g768(64, 6), g3072(64, 24);

  transposeW<<<(int)((SZ_W  + 255) / 256), blk, 0, stream>>>(Wq,  WTq,  769,  KP1, DD);
  transposeW<<<(int)((SZ_W  + 255) / 256), blk, 0, stream>>>(Wk,  WTk,  769,  KP1, DD);
  transposeW<<<(int)((SZ_W  + 255) / 256), blk, 0, stream>>>(Wv,  WTv,  769,  KP1, DD);
  transposeW<<<(int)((SZ_W  + 255) / 256), blk, 0, stream>>>(Wo,  WTo,  769,  KP1, DD);
  transposeW<<<(int)((SZ_M1 + 255) / 256), blk, 0, stream>>>(Wm1, WTm1, 769,  KP1, DM);
  transposeW<<<(int)((SZ_M2 + 255) / 256), blk, 0, stream>>>(Wm2, WTm2, 3073, KP2, DD);

  ln1_kernel<<<16, blk, 0, stream>>>(x, g1, b1, xlnT);

  gemm_kt<<<g768, blk, 0, stream>>>(WTq, xlnT, qT, KP1, DD, 0, nullptr);
  gemm_kt<<<g768, blk, 0, stream>>>(WTk, xlnT, kT, KP1, DD, 0, nullptr);
  gemm_kt<<<g768, blk, 0, stream>>>(WTv, xlnT, vT, KP1, DD, 0, nullptr);

  head_prep<<<192, blk, 0, stream>>>(qT, kT, vT, qt_, kt_, vtanJ);

  attn_kernel<<<dim3(8, 48), blk, 0, stream>>>(qT, kT, qt_, kt_, vtanJ, yT, tH, ar, tr, lr);

  tnew_kernel<<<16, blk, 0, stream>>>(tH, yT);

  gemm_kt<<<g768, blk, 0, stream>>>(WTo, yT, outT, KP1, DD, 2, x);

  ln2_kernel<<<16, blk, 0, stream>>>(outT, g2, b2, hT);

  gemm_kt<<<g3072, blk, 0, stream>>>(WTm1, hT, mT + T4, KP1, DM, 1, nullptr);

  mtime_kernel<<<16, blk, 0, stream>>>(mT);

  gemm_kt<<<g768, blk, 0, stream>>>(WTm2, mT, finT, KP2, DD, 3, outT);

  final_kernel<<<16, blk, 0, stream>>>(finT, out);
}